// MiniMaxM2Attention_78752520339603
// MI455X (gfx1250) — compile-verified
//
#include <hip/hip_runtime.h>
#include <cstdint>

// Fused QK RMSNorm for MI455X (gfx1250, wave32).
// Bandwidth-bound: ~940 MB of HBM traffic -> ~40us floor at 23.3 TB/s.
// One-pass per row: global -> LDS via gfx1250 async-to-LDS ops (ASYNCcnt),
// block reduction for sum-of-squares, then scale + b128 store.

typedef __attribute__((ext_vector_type(4))) float f4;
typedef __attribute__((ext_vector_type(4))) int   v4i;

#define RMS_EPS 1e-6f

// ---- gfx1250 async global->LDS copy (ASYNCcnt path), with safe fallback ----
__device__ __forceinline__ void async_copy_b128(const float* g, float* l) {
#if __has_builtin(__builtin_amdgcn_global_load_async_to_lds_b128)
  typedef __attribute__((address_space(1))) v4i* gp_t;
  typedef __attribute__((address_space(3))) v4i* lp_t;
  __builtin_amdgcn_global_load_async_to_lds_b128(
      (gp_t)(v4i*)g, (lp_t)(v4i*)l, /*imm offset=*/0, /*cpol=*/0);
#else
  *(f4*)l = *(const f4*)g;  // fallback: sync copy through VGPRs
#endif
}

__device__ __forceinline__ void wait_async0() {
#if __has_builtin(__builtin_amdgcn_s_wait_asynccnt)
  __builtin_amdgcn_s_wait_asynccnt(0);
#else
  asm volatile("s_wait_asynccnt 0" ::: "memory");
#endif
}

// One workgroup (256 threads = 8 wave32) per row.
// D = row length in floats, VPT = float4 chunks per thread, D == VPT*256*4.
template <int D, int VPT>
__launch_bounds__(256)
__global__ void rmsnorm_rows(const float* __restrict__ x,
                             const float* __restrict__ w,
                             float* __restrict__ out) {
  static_assert(D == VPT * 256 * 4, "row must tile exactly");
  __shared__ float srow[D];      // row staging (24KB for q, 4KB for k)
  __shared__ float sred[9];      // 8 wave partials + broadcast inv_rms

  const int t = threadIdx.x;
  const size_t row = blockIdx.x;
  const float* __restrict__ xr = x + row * (size_t)D;
  float* __restrict__ orow = out + row * (size_t)D;

  // 1) Stream the row into LDS with async b128 copies (fully coalesced).
#pragma unroll
  for (int i = 0; i < VPT; ++i) {
    const int c4 = (t + i * 256) * 4;
    async_copy_b128(xr + c4, srow + c4);
  }
  wait_async0();          // this wave's async copies complete
  __syncthreads();        // ordering fence across the 8 waves (cheap)

  // 2) Read back own chunks, accumulate sum of squares in fp32 FMAs.
  f4 v[VPT];
  float ss = 0.0f;
#pragma unroll
  for (int i = 0; i < VPT; ++i) {
    const int c4 = (t + i * 256) * 4;
    v[i] = *(const f4*)(srow + c4);
    ss = fmaf(v[i].x, v[i].x, ss);
    ss = fmaf(v[i].y, v[i].y, ss);
    ss = fmaf(v[i].z, v[i].z, ss);
    ss = fmaf(v[i].w, v[i].w, ss);
  }

  // 3) wave32 butterfly reduce, then 8-wave LDS reduce.
#pragma unroll
  for (int off = 16; off > 0; off >>= 1)
    ss += __shfl_xor(ss, off, 32);
  if ((t & 31) == 0) sred[t >> 5] = ss;
  __syncthreads();
  if (t == 0) {
    float tot = sred[0];
#pragma unroll
    for (int i = 1; i < 8; ++i) tot += sred[i];
    sred[8] = rsqrtf(tot * (1.0f / (float)D) + RMS_EPS);
  }
  __syncthreads();
  const float inv = sred[8];

  // 4) Scale by inv_rms * weight and store b128 (weights are L2-resident).
#pragma unroll
  for (int i = 0; i < VPT; ++i) {
    const int c4 = (t + i * 256) * 4;
    const f4 wv = *(const f4*)(w + c4);
    f4 o;
    o.x = v[i].x * inv * wv.x;
    o.y = v[i].y * inv * wv.y;
    o.z = v[i].z * inv * wv.z;
    o.w = v[i].w * inv * wv.w;
    *(f4*)(orow + c4) = o;
  }
}

extern "C" void kernel_launch(void* const* d_in, const int* in_sizes, int n_in,
                              void* d_out, int out_size, void* d_ws, size_t ws_size,
                              hipStream_t stream) {
  (void)in_sizes; (void)n_in; (void)out_size; (void)d_ws; (void)ws_size;

  constexpr int Bn = 16384;
  constexpr int D1 = 6144;
  constexpr int D2 = 1024;

  const float* q  = (const float*)d_in[0];
  const float* k  = (const float*)d_in[1];
  const float* qw = (const float*)d_in[2];
  const float* kw = (const float*)d_in[3];
  float* out_q = (float*)d_out;
  float* out_k = out_q + (size_t)Bn * D1;

  rmsnorm_rows<D1, 6><<<Bn, 256, 0, stream>>>(q, qw, out_q);
  rmsnorm_rows<D2, 1><<<Bn, 256, 0, stream>>>(k, kw, out_k);
}